// OverlapTripletLoss05_11991548690927
// MI455X (gfx1250) — compile-verified
//
#include <hip/hip_runtime.h>
#include <hip/hip_bf16.h>
#include <math.h>

typedef __attribute__((ext_vector_type(2))) float v2f;
typedef __attribute__((ext_vector_type(8))) float v8f;

#define C_CLS 64
#define K_SEL 64
#define DIMS  256
#define ALPHA 1.0f
#define SORT_N 2048
#define CHUNK (SORT_N - K_SEL)

// ---------------- init: mu norms + zero counters ----------------
__global__ void init_kernel(const float* __restrict__ mu, float* __restrict__ muN,
                            int* __restrict__ counts, int* __restrict__ cursors) {
    int t = threadIdx.x;                 // 128 threads
    if (t < C_CLS) {
        const float* r = mu + (size_t)t * DIMS;
        float s = 0.f;
        for (int k = 0; k < DIMS; ++k) s += r[k] * r[k];
        muN[t] = s;
    } else {
        counts[t - C_CLS]  = 0;
        cursors[t - C_CLS] = 0;
    }
}

// ---------------- GEMM (f32 WMMA 16x16x4) + distances ----------------
// Block = 128 threads = 4 waves. Each block: 16-column tile of x, all 64 mu rows.
// Wave w computes rows [16w, 16w+16). x tile staged in LDS (padded stride).
__global__ __launch_bounds__(128) void gemm_dist_kernel(
    const float* __restrict__ x, const float* __restrict__ mu,
    const float* __restrict__ muN, float* __restrict__ Dmat, int Ntot) {
    __shared__ float xs[16 * 260];   // 16 rows, stride 260 (bank-conflict pad)
    __shared__ float xn[16];

    const int tid = threadIdx.x;
    const int n0  = blockIdx.x * 16;

    // Coalesced load of x tile: 1024 float4s across 128 threads
    for (int f = tid; f < 1024; f += 128) {
        int row = f >> 6;            // 64 float4 per 256-wide row
        int k4  = f & 63;
        float4 v = ((const float4*)(x + (size_t)(n0 + row) * DIMS))[k4];
        float* dst = &xs[row * 260 + k4 * 4];
        dst[0] = v.x; dst[1] = v.y; dst[2] = v.z; dst[3] = v.w;
    }
    __syncthreads();
    if (tid < 16) {                  // per-column squared norms
        float s = 0.f;
        const float* r = &xs[tid * 260];
        for (int k = 0; k < DIMS; ++k) s += r[k] * r[k];
        xn[tid] = s;
    }
    __syncthreads();

    const int wave = tid >> 5;       // row tile 0..3
    const int lane = tid & 31;
    const int rc   = lane & 15;      // A row index == B col index for this lane
    const int kb   = (lane >> 4) * 2;// K sub-offset per ISA 16x4 f32 layout
    const int rowbase = wave * 16;

    const float* mrow = mu + (size_t)(rowbase + rc) * DIMS + kb;
    const float* xcol = &xs[rc * 260 + kb];

    v8f acc = {};
    for (int k = 0; k < DIMS; k += 4) {
        v2f a; a.x = mrow[k]; a.y = mrow[k + 1];
        v2f b; b.x = xcol[k]; b.y = xcol[k + 1];
        acc = __builtin_amdgcn_wmma_f32_16x16x4_f32(
            /*neg_a*/false, a, /*neg_b*/false, b,
            /*c_mod*/(short)0, acc, /*reuse_a*/false, /*reuse_b*/false);
    }

    // C/D layout: M = v + 8*(lane>>4), N = lane&15
    const int ncol  = n0 + rc;
    const float xnv = xn[rc];
    const int mhalf = (lane >> 4) * 8;
#pragma unroll
    for (int v = 0; v < 8; ++v) {
        int M = rowbase + mhalf + v;
        float d2 = muN[M] + xnv - 2.0f * acc[v];
        Dmat[(size_t)M * Ntot + ncol] = sqrtf(fmaxf(d2, 0.0f));
    }
}

// ---------------- counting sort of indices by label ----------------
__global__ void count_kernel(const int* __restrict__ y, int* __restrict__ counts, int N) {
    int n = blockIdx.x * blockDim.x + threadIdx.x;
    if (n < N) atomicAdd(&counts[y[3 * n + 2] & (C_CLS - 1)], 1);
}

__global__ void scan_kernel(const int* __restrict__ counts, int* __restrict__ offsets) {
    if (threadIdx.x == 0) {
        int s = 0;
        for (int i = 0; i < C_CLS; ++i) { offsets[i] = s; s += counts[i]; }
    }
}

__global__ void scatter_kernel(const int* __restrict__ y, const int* __restrict__ offsets,
                               int* __restrict__ cursors, int* __restrict__ lists, int N) {
    int n = blockIdx.x * blockDim.x + threadIdx.x;
    if (n < N) {
        int lab = y[3 * n + 2] & (C_CLS - 1);
        int p = atomicAdd(&cursors[lab], 1);
        lists[offsets[lab] + p] = n;
    }
}

// ---------------- bitonic sort of 2048 floats in LDS (ascending) ----------------
__device__ __forceinline__ void bitonic2048(float* buf) {
    for (unsigned k = 2; k <= SORT_N; k <<= 1) {
        for (unsigned j = k >> 1; j > 0; j >>= 1) {
            __syncthreads();
            for (unsigned t = threadIdx.x; t < SORT_N / 2; t += blockDim.x) {
                unsigned i   = 2u * t - (t & (j - 1u));
                unsigned ixj = i + j;
                bool up = ((i & k) == 0u);
                float a = buf[i], b = buf[ixj];
                if ((a > b) == up) { buf[i] = b; buf[ixj] = a; }
            }
        }
    }
    __syncthreads();
}

// ---------------- top-K selection per (c, m) cell ----------------
// off-diagonal: mean of K smallest d  -> negM[m*64+c]
// diagonal:     mean of K largest d   -> posmean[c]   (via sign flip)
__global__ __launch_bounds__(256) void select_kernel(
    const float* __restrict__ Dmat, const int* __restrict__ lists,
    const int* __restrict__ counts, const int* __restrict__ offsets,
    float* __restrict__ negM, float* __restrict__ posmean, int Ntot) {
    __shared__ float buf[SORT_N];
    const int cell = blockIdx.x;
    const int m = cell >> 6, c = cell & (C_CLS - 1);
    const float sgn = (m == c) ? -1.0f : 1.0f;
    const int cnt = counts[m], off = offsets[m];
    const float* Drow = Dmat + (size_t)c * Ntot;

    if (threadIdx.x < K_SEL) buf[threadIdx.x] = INFINITY;   // running best

    int processed = 0;
    while (processed < cnt) {
        int chunk = min(cnt - processed, CHUNK);
        __syncthreads();
        for (int i = threadIdx.x; i < CHUNK; i += blockDim.x) {
            float v = INFINITY;
            if (i < chunk) v = sgn * Drow[lists[off + processed + i]];
            buf[K_SEL + i] = v;
        }
        __syncthreads();
        bitonic2048(buf);        // best K land in buf[0..K_SEL)
        processed += chunk;
    }
    __syncthreads();

    if (threadIdx.x == 0) {
        float s = 0.f;
        for (int i = 0; i < K_SEL; ++i) s += buf[i];
        float mean = s * (1.0f / K_SEL);
        if (m == c) posmean[c] = -mean;          // flip back: mean of largest d
        else        negM[m * C_CLS + c] = mean;  // mean of smallest d
    }
}

// ---------------- hinge + off-diagonal reduction ----------------
__global__ __launch_bounds__(256) void finalize_kernel(
    const float* __restrict__ negM, const float* __restrict__ posmean,
    float* __restrict__ out, float invN) {
    __shared__ float red[256];
    float s = 0.f;
    for (int i = threadIdx.x; i < C_CLS * C_CLS; i += 256) {
        int m = i >> 6, c = i & (C_CLS - 1);
        if (m != c) {
            float h = ALPHA + posmean[c] - negM[i];
            s += (h > 0.f) ? h : 0.f;
        }
    }
    red[threadIdx.x] = s;
    __syncthreads();
    for (int w = 128; w > 0; w >>= 1) {
        if (threadIdx.x < w) red[threadIdx.x] += red[threadIdx.x + w];
        __syncthreads();
    }
    if (threadIdx.x == 0) out[0] = red[0] * invN;
}

// ---------------- host launcher ----------------
extern "C" void kernel_launch(void* const* d_in, const int* in_sizes, int n_in,
                              void* d_out, int out_size, void* d_ws, size_t ws_size,
                              hipStream_t stream) {
    const float* x  = (const float*)d_in[0];
    const int*   y  = (const int*)d_in[1];
    const float* mu = (const float*)d_in[2];
    float* out = (float*)d_out;

    const int N = in_sizes[0] / DIMS;     // 131072

    // workspace layout
    char* ws = (char*)d_ws;
    float* muN     = (float*)(ws + 0);                    // 64 f
    int*   counts  = (int*)  (ws + 256);                  // 64 i
    int*   offsets = (int*)  (ws + 512);                  // 64 i
    int*   cursors = (int*)  (ws + 768);                  // 64 i
    float* posmean = (float*)(ws + 1024);                 // 64 f
    float* negM    = (float*)(ws + 1280);                 // 4096 f
    float* Dmat    = (float*)(ws + 17664);                // 64*N f (32 MB)
    int*   lists   = (int*)  (ws + 17664 + (size_t)C_CLS * N * sizeof(float)); // N i

    // 1. mu norms + zero counters
    init_kernel<<<1, 128, 0, stream>>>(mu, muN, counts, cursors);

    // 2. WMMA GEMM + distances
    gemm_dist_kernel<<<N / 16, 128, 0, stream>>>(x, mu, muN, Dmat, N);

    // 3. counting sort of indices by label
    int g = (N + 255) / 256;
    count_kernel<<<g, 256, 0, stream>>>(y, counts, N);
    scan_kernel<<<1, 64, 0, stream>>>(counts, offsets);
    scatter_kernel<<<g, 256, 0, stream>>>(y, offsets, cursors, lists, N);

    // 4. per-cell top-K selection (4096 cells)
    select_kernel<<<C_CLS * C_CLS, 256, 0, stream>>>(Dmat, lists, counts, offsets,
                                                     negM, posmean, N);

    // 5. hinge + reduction -> scalar
    finalize_kernel<<<1, 256, 0, stream>>>(negM, posmean, out, 1.0f / (float)N);
}